// DGBasedVonMisesFisherKLD_34299608826309
// MI455X (gfx1250) — compile-verified
//
#include <hip/hip_runtime.h>
#include <hip/hip_bf16.h>
#include <math.h>

// D = 32 (DIM_Z) is fixed by the reference model.
#define DIMZ 32
#define WAVES_PER_BLOCK 8
#define JTILE 16
#define STAGE_ROWS 64   // i-rows staged into LDS per step (8 KB)

typedef __attribute__((ext_vector_type(2))) float v2f;
typedef __attribute__((ext_vector_type(8))) float v8f;

// ---------------------------------------------------------------------------
// Kernel 1: A[i] = kappa * mu[i] / ||mu[i]||  (one wave32 per row), and zero
// the global log-sum accumulator (stream order guarantees it's ready).
// ---------------------------------------------------------------------------
__global__ void vmf_prep(const float* __restrict__ mu,
                         const float* __restrict__ kappa_p,
                         float* __restrict__ A,
                         float* __restrict__ accum, int B) {
  if (blockIdx.x == 0 && threadIdx.x == 0) *accum = 0.0f;
  const int tid  = blockIdx.x * blockDim.x + threadIdx.x;
  const int row  = tid >> 5;
  const int lane = tid & 31;
  if (row >= B) return;
  float v  = mu[(size_t)row * DIMZ + lane];
  float ss = v * v;
  ss += __shfl_xor(ss, 1, 32);
  ss += __shfl_xor(ss, 2, 32);
  ss += __shfl_xor(ss, 4, 32);
  ss += __shfl_xor(ss, 8, 32);
  ss += __shfl_xor(ss, 16, 32);
  const float scale = kappa_p[0] * rsqrtf(ss);
  A[(size_t)row * DIMZ + lane] = v * scale;
}

// ---------------------------------------------------------------------------
// Kernel 2: fused GEMM (kappa*mu_n · z^T) + per-column sum of exp(acc - kappa)
// via V_WMMA_F32_16X16X4_F32, accumulating sum_j log(s_j) globally.
//
// WMMA 16x16x4 f32 fragment layouts (wave32):
//   A (16x4):  lanes 0-15 -> M=lane, {v0=K0, v1=K1}; lanes 16-31 -> {K2, K3}
//   B (4x16):  lanes 0-15 -> N=lane, {v0=K0, v1=K1}; lanes 16-31 -> {K2, K3}
//   C (16x16): lane<16: (M=v, N=lane); lane>=16: (M=v+8, N=lane-16)
// So each lane reads float2 pairs at stride 4 floats from its row.
// ---------------------------------------------------------------------------
__global__ void __launch_bounds__(256)
vmf_main(const float* __restrict__ A, const float* __restrict__ z,
         const float* __restrict__ kappa_p, float* __restrict__ accum,
         int B, int Jtotal) {
  __shared__ float Alds[STAGE_ROWS * DIMZ];

  const int tid   = threadIdx.x;
  const int wave  = tid >> 5;
  const int lane  = tid & 31;
  const int half  = lane >> 4;     // 0: K lanes {0,1}, 1: K lanes {2,3}
  const int lr    = lane & 15;
  const int jbase = (blockIdx.x * WAVES_PER_BLOCK + wave) * JTILE;
  const float kappa = kappa_p[0];

  // B fragments: this wave's 16 z-columns, kept in registers for the whole
  // i-sweep (16 VGPRs).
  v2f bfrag[8];
  {
    const float* zr = z + (size_t)(jbase + lr) * DIMZ + 2 * half;
#pragma unroll
    for (int kk = 0; kk < 8; ++kk) {
      bfrag[kk].x = zr[4 * kk + 0];
      bfrag[kk].y = zr[4 * kk + 1];
    }
  }

  float ssum = 0.0f;  // per-lane partial of s_j (half the M rows)

  for (int ibase = 0; ibase < B; ibase += STAGE_ROWS) {
    __syncthreads();
    {  // cooperative stage: 64 rows x 32 floats = 8 KB via float4
      const float4* src = (const float4*)(A + (size_t)ibase * DIMZ);
      float4* dst = (float4*)Alds;
#pragma unroll
      for (int t = tid; t < STAGE_ROWS * DIMZ / 4; t += 256) dst[t] = src[t];
    }
    __syncthreads();

#pragma unroll
    for (int sib = 0; sib < STAGE_ROWS / 16; ++sib) {
      const float* ar = Alds + (sib * 16 + lr) * DIMZ + 2 * half;
      v8f c = {0.f, 0.f, 0.f, 0.f, 0.f, 0.f, 0.f, 0.f};
#pragma unroll
      for (int kk = 0; kk < 8; ++kk) {
        v2f a;
        a.x = ar[4 * kk + 0];
        a.y = ar[4 * kk + 1];
        // D = A x B + C   (full f32 path)
        c = __builtin_amdgcn_wmma_f32_16x16x4_f32(
            /*neg_a=*/false, a, /*neg_b=*/false, bfrag[kk],
            /*c_mod=*/(short)0, c, /*reuse_a=*/false, /*reuse_b=*/false);
      }
      // c[v] = kappa * <mu_n_i, z_j>; dot <= 1 so exp(c - kappa) never
      // overflows and the dominant term (~e^-13) never underflows.
#pragma unroll
      for (int v = 0; v < 8; ++v) ssum += __expf(c[v] - kappa);
    }
  }

  // Combine M halves: lanes l and l+16 hold complementary row sums of col l%16.
  ssum += __shfl_xor(ssum, 16, 32);
  float lg = __logf(ssum);  // log s_j for column j = jbase + (lane & 15)
  // Sum the 16 column logs (butterfly within each 16-lane half; halves equal).
  lg += __shfl_xor(lg, 1, 32);
  lg += __shfl_xor(lg, 2, 32);
  lg += __shfl_xor(lg, 4, 32);
  lg += __shfl_xor(lg, 8, 32);
  if (lane == 0) atomicAdd(accum, lg);
}

// ---------------------------------------------------------------------------
// Kernel 3: okl = mean_j(kappa + log s_j + log_C_kappa - log B) - log_C_zero
// ---------------------------------------------------------------------------
__global__ void vmf_final(const float* __restrict__ accum,
                          const float* __restrict__ kappa_p,
                          const float* __restrict__ logCk_p,
                          const float* __restrict__ logC0_p,
                          float* __restrict__ out, int B, int Jtotal) {
  if (threadIdx.x == 0 && blockIdx.x == 0) {
    const float mean_log = accum[0] / (float)Jtotal;
    out[0] = mean_log + logCk_p[0] + kappa_p[0] - __logf((float)B) - logC0_p[0];
  }
}

// ---------------------------------------------------------------------------
// Launch. Inputs: mu [B*32], z [B*n*32], kappa[1], log_C_kappa[1],
// log_C_zero[1], n_samples[1]. Output: scalar float.
// Workspace: A_scaled (B*32 floats) + 1 accumulator float.
// ---------------------------------------------------------------------------
extern "C" void kernel_launch(void* const* d_in, const int* in_sizes, int n_in,
                              void* d_out, int out_size, void* d_ws, size_t ws_size,
                              hipStream_t stream) {
  const float* mu    = (const float*)d_in[0];
  const float* z     = (const float*)d_in[1];
  const float* kappa = (const float*)d_in[2];
  const float* logCk = (const float*)d_in[3];
  const float* logC0 = (const float*)d_in[4];

  const int B      = in_sizes[0] / DIMZ;  // 2048
  const int Jtotal = in_sizes[1] / DIMZ;  // 65536

  float* A     = (float*)d_ws;            // B*32 floats
  float* accum = A + (size_t)B * DIMZ;    // 1 float
  float* out   = (float*)d_out;

  vmf_prep<<<(B * 32 + 255) / 256, 256, 0, stream>>>(mu, kappa, A, accum, B);
  vmf_main<<<Jtotal / (WAVES_PER_BLOCK * JTILE), 256, 0, stream>>>(
      A, z, kappa, accum, B, Jtotal);
  vmf_final<<<1, 1, 0, stream>>>(accum, kappa, logCk, logC0, out, B, Jtotal);
}